// GATT_34445637714178
// MI455X (gfx1250) — compile-verified
//
#include <hip/hip_runtime.h>
#include <math.h>

typedef float v2f __attribute__((ext_vector_type(2)));
typedef float v8f __attribute__((ext_vector_type(8)));

#define DIN 128
#define C1  64
#define NEG_SLOPE 0.2f

// ---------- helpers ----------
__device__ __forceinline__ float atomicMaxF(float* addr, float val) {
  // order-preserving integer mapping for IEEE-754 floats
  if (val >= 0.0f)
    return __int_as_float(atomicMax((int*)addr, __float_as_int(val)));
  else
    return __uint_as_float(atomicMin((unsigned int*)addr, __float_as_uint(val)));
}

// ---------- init ----------
__global__ void k_init(float* __restrict__ out1, float* __restrict__ m,
                       float* __restrict__ z, float* __restrict__ outAcc, int n) {
  int i = blockIdx.x * blockDim.x + threadIdx.x;
  if (i < n * C1) out1[i] = 0.0f;
  if (i < n) { m[i] = -INFINITY; z[i] = 0.0f; outAcc[i] = 0.0f; }
}

// ---------- layer1 GEMM: h[N,64] = x[N,128] @ W1[128,64] via fp32 WMMA ----------
// One wave -> 16 rows x 64 cols tile. K stepped by 4 with V_WMMA_F32_16X16X4_F32.
__global__ __launch_bounds__(256) void k_gemm_wmma(const float* __restrict__ x,
                                                   const float* __restrict__ W,
                                                   float* __restrict__ h, int n) {
  __shared__ float Wt[C1 * DIN];  // W transposed: Wt[col][k]
  int tid = threadIdx.x;
#pragma unroll
  for (int it = 0; it < (DIN * C1) / 256; ++it) {
    int idx = tid + it * 256;
    int k  = idx >> 6;   // row of W (0..127)
    int nn = idx & 63;   // col of W (0..63)
    Wt[nn * DIN + k] = W[idx];
  }
  __syncthreads();

  int wave = tid >> 5;
  int lane = tid & 31;
  int half = lane >> 4;   // 0: K=0,1  1: K=2,3  (A/B frag layout)
  int l    = lane & 15;
  int rowBase = blockIdx.x * 128 + wave * 16;

  int arow = rowBase + l;
  if (arow > n - 1) arow = n - 1;              // clamp; masked at store
  const float* ap = x + (size_t)arow * DIN + 2 * half;

  v8f c0 = {}, c1 = {}, c2 = {}, c3 = {};
#pragma unroll 4
  for (int k0 = 0; k0 < DIN; k0 += 4) {
    v2f a = *(const v2f*)(ap + k0);
    const float* wp = &Wt[l * DIN + k0 + 2 * half];
    v2f b0 = *(const v2f*)(wp);
    v2f b1 = *(const v2f*)(wp + 16 * DIN);
    v2f b2 = *(const v2f*)(wp + 32 * DIN);
    v2f b3 = *(const v2f*)(wp + 48 * DIN);
    c0 = __builtin_amdgcn_wmma_f32_16x16x4_f32(false, a, false, b0, (short)0, c0, false, false);
    c1 = __builtin_amdgcn_wmma_f32_16x16x4_f32(false, a, false, b1, (short)0, c1, false, false);
    c2 = __builtin_amdgcn_wmma_f32_16x16x4_f32(false, a, false, b2, (short)0, c2, false, false);
    c3 = __builtin_amdgcn_wmma_f32_16x16x4_f32(false, a, false, b3, (short)0, c3, false, false);
  }

  // C/D layout: VGPR v, lanes 0-15 -> row rowBase+v, lanes 16-31 -> row rowBase+8+v; col = ntile*16 + l
#pragma unroll
  for (int v = 0; v < 8; ++v) {
    int row = rowBase + 8 * half + v;
    if (row < n) {
      float* hp = h + (size_t)row * C1 + l;
      hp[0]  = c0[v];
      hp[16] = c1[v];
      hp[32] = c2[v];
      hp[48] = c3[v];
    }
  }
}

// ---------- per-node attention logits: e_src = h.a_src, e_dst = h.a_dst ----------
__global__ void k_logits(const float* __restrict__ h, const float* __restrict__ a_src,
                         const float* __restrict__ a_dst, float* __restrict__ esrc,
                         float* __restrict__ edst, int n) {
  int i = blockIdx.x * blockDim.x + threadIdx.x;
  if (i >= n) return;
  const float4* hp = (const float4*)(h + (size_t)i * C1);
  float s = 0.0f, d = 0.0f;
#pragma unroll
  for (int c = 0; c < C1 / 4; ++c) {
    float4 hv = hp[c];
    float4 as = ((const float4*)a_src)[c];
    float4 ad = ((const float4*)a_dst)[c];
    s += hv.x * as.x + hv.y * as.y + hv.z * as.z + hv.w * as.w;
    d += hv.x * ad.x + hv.y * ad.y + hv.z * ad.z + hv.w * ad.w;
  }
  esrc[i] = s;
  edst[i] = d;
}

// ---------- edge pass 1: leaky-relu logit + segment max ----------
__global__ void k_edge_max(const int* __restrict__ ei, int E, int n,
                           const float* __restrict__ esrc, const float* __restrict__ edst,
                           float* __restrict__ m, float* __restrict__ eedge) {
  int idx = blockIdx.x * blockDim.x + threadIdx.x;
  int Etot = E + n;
  if (idx >= Etot) return;
  int s, d;
  if (idx < E) { s = ei[idx]; d = ei[E + idx]; }
  else         { s = d = idx - E; }             // self-loop
  float v = esrc[s] + edst[d];
  v = v > 0.0f ? v : NEG_SLOPE * v;
  eedge[idx] = v;
  atomicMaxF(&m[d], v);
}

// ---------- edge pass 2: p = exp(e - m[dst]); z[dst] += p; eedge <- p ----------
__global__ void k_edge_expsum(const int* __restrict__ ei, int E, int n,
                              const float* __restrict__ m, float* __restrict__ z,
                              float* __restrict__ eedge) {
  int idx = blockIdx.x * blockDim.x + threadIdx.x;
  int Etot = E + n;
  if (idx >= Etot) return;
  int d = (idx < E) ? ei[E + idx] : (idx - E);
  float p = __expf(eedge[idx] - m[d]);
  eedge[idx] = p;
  atomicAdd(&z[d], p);
}

// ---------- edge pass 3 (layer 1): out1[dst] += (p/z[dst]) * h[src], 64 channels ----------
// 16 lanes per edge, float4 per lane -> fully coalesced gather of h[src]
__global__ void k_edge_aggr64(const int* __restrict__ ei, int E, int n,
                              const float* __restrict__ z, const float* __restrict__ eedge,
                              const float* __restrict__ h, float* __restrict__ out1) {
  long long t = (long long)blockIdx.x * blockDim.x + threadIdx.x;
  int idx = (int)(t >> 4);
  int sub = (int)(t & 15);
  int Etot = E + n;
  if (idx >= Etot) return;
  int s, d;
  if (idx < E) { s = ei[idx]; d = ei[E + idx]; }
  else         { s = d = idx - E; }
  float alpha = eedge[idx] / z[d];
  float4 hv = *(const float4*)(h + (size_t)s * C1 + sub * 4);
  float* op = out1 + (size_t)d * C1 + sub * 4;
  atomicAdd(op + 0, alpha * hv.x);
  atomicAdd(op + 1, alpha * hv.y);
  atomicAdd(op + 2, alpha * hv.z);
  atomicAdd(op + 3, alpha * hv.w);
}

// ---------- layer-2 prep: g = relu(out1+b1)@W2 (scalar), logits, reset m/z ----------
__global__ void k_layer2_prep(const float* __restrict__ out1, const float* __restrict__ b1,
                              const float* __restrict__ W2, const float* __restrict__ a_src2,
                              const float* __restrict__ a_dst2, float* __restrict__ g,
                              float* __restrict__ esrc, float* __restrict__ edst,
                              float* __restrict__ m, float* __restrict__ z, int n) {
  int i = blockIdx.x * blockDim.x + threadIdx.x;
  if (i >= n) return;
  const float4* op = (const float4*)(out1 + (size_t)i * C1);
  float acc = 0.0f;
#pragma unroll
  for (int c = 0; c < C1 / 4; ++c) {
    float4 v = op[c];
    float4 bb = ((const float4*)b1)[c];
    float4 ww = ((const float4*)W2)[c];
    float v0 = fmaxf(v.x + bb.x, 0.0f);
    float v1 = fmaxf(v.y + bb.y, 0.0f);
    float v2 = fmaxf(v.z + bb.z, 0.0f);
    float v3 = fmaxf(v.w + bb.w, 0.0f);
    acc += v0 * ww.x + v1 * ww.y + v2 * ww.z + v3 * ww.w;
  }
  g[i] = acc;
  esrc[i] = acc * a_src2[0];
  edst[i] = acc * a_dst2[0];
  m[i] = -INFINITY;
  z[i] = 0.0f;
}

// ---------- edge pass 3 (layer 2): scalar aggregate ----------
__global__ void k_edge_aggr1(const int* __restrict__ ei, int E, int n,
                             const float* __restrict__ z, const float* __restrict__ eedge,
                             const float* __restrict__ g, float* __restrict__ outAcc) {
  int idx = blockIdx.x * blockDim.x + threadIdx.x;
  int Etot = E + n;
  if (idx >= Etot) return;
  int s, d;
  if (idx < E) { s = ei[idx]; d = ei[E + idx]; }
  else         { s = d = idx - E; }
  float alpha = eedge[idx] / z[d];
  atomicAdd(&outAcc[d], alpha * g[s]);
}

// ---------- final: sigmoid(outAcc + b2) in place ----------
__global__ void k_final(float* __restrict__ out, const float* __restrict__ b2, int n) {
  int i = blockIdx.x * blockDim.x + threadIdx.x;
  if (i >= n) return;
  float v = out[i] + b2[0];
  out[i] = 1.0f / (1.0f + __expf(-v));
}

extern "C" void kernel_launch(void* const* d_in, const int* in_sizes, int n_in,
                              void* d_out, int out_size, void* d_ws, size_t ws_size,
                              hipStream_t stream) {
  const int*   ei     = (const int*)d_in[0];
  const float* x      = (const float*)d_in[1];
  const float* W1     = (const float*)d_in[2];
  const float* a_src1 = (const float*)d_in[3];
  const float* a_dst1 = (const float*)d_in[4];
  const float* b1     = (const float*)d_in[5];
  const float* W2     = (const float*)d_in[6];
  const float* a_src2 = (const float*)d_in[7];
  const float* a_dst2 = (const float*)d_in[8];
  const float* b2     = (const float*)d_in[9];

  const int E = in_sizes[0] / 2;
  const int n = in_sizes[1] / DIN;
  const int Etot = E + n;

  float* f     = (float*)d_ws;
  float* h     = f;                          // [n*64]
  float* out1  = f + (size_t)64 * n;         // [n*64]
  float* esrc  = f + (size_t)128 * n;        // [n]
  float* edst  = f + (size_t)129 * n;        // [n]
  float* m     = f + (size_t)130 * n;        // [n]
  float* z     = f + (size_t)131 * n;        // [n]
  float* g     = f + (size_t)132 * n;        // [n]
  float* eedge = f + (size_t)133 * n;        // [Etot]
  float* outAcc = (float*)d_out;             // [n] accumulator -> sigmoid in place

  const int B = 256;
  dim3 blk(B);

  // init
  k_init<<<dim3((n * C1 + B - 1) / B), blk, 0, stream>>>(out1, m, z, outAcc, n);

  // layer 1
  k_gemm_wmma<<<dim3((n + 127) / 128), blk, 0, stream>>>(x, W1, h, n);
  k_logits<<<dim3((n + B - 1) / B), blk, 0, stream>>>(h, a_src1, a_dst1, esrc, edst, n);
  k_edge_max<<<dim3((Etot + B - 1) / B), blk, 0, stream>>>(ei, E, n, esrc, edst, m, eedge);
  k_edge_expsum<<<dim3((Etot + B - 1) / B), blk, 0, stream>>>(ei, E, n, m, z, eedge);
  {
    long long tthreads = (long long)Etot * 16;
    int grid = (int)((tthreads + B - 1) / B);
    k_edge_aggr64<<<dim3(grid), blk, 0, stream>>>(ei, E, n, z, eedge, h, out1);
  }

  // layer 2 (scalar channel)
  k_layer2_prep<<<dim3((n + B - 1) / B), blk, 0, stream>>>(out1, b1, W2, a_src2, a_dst2,
                                                           g, esrc, edst, m, z, n);
  k_edge_max<<<dim3((Etot + B - 1) / B), blk, 0, stream>>>(ei, E, n, esrc, edst, m, eedge);
  k_edge_expsum<<<dim3((Etot + B - 1) / B), blk, 0, stream>>>(ei, E, n, m, z, eedge);
  k_edge_aggr1<<<dim3((Etot + B - 1) / B), blk, 0, stream>>>(ei, E, n, z, eedge, g, outAcc);

  k_final<<<dim3((n + B - 1) / B), blk, 0, stream>>>(outAcc, b2, n);
}